// Gate_Net_46703474377219
// MI455X (gfx1250) — compile-verified
//
#include <hip/hip_runtime.h>
#include <stdint.h>

// Reference constants: D=512, L=258, n = L-2 = 256, r = 1..255.
#define D_ROWS 512
#define L_LEN  258
#define N_SC   256
#define N_R    255
#define PAD    256   // zero-padding on each side of the LDS row

typedef __attribute__((ext_vector_type(2))) float v2f;

// Store-bandwidth-bound kernel: 267 MB f32 out @ 23.3 TB/s => ~11.5 us floor.
// One block per row d, 128 threads (4 wave32), each thread owns 2 adjacent
// columns => b64 NT stores (256 B per wave store).
// Gate folded to 1 clamped-FMA + 1 mul per element; OOB handled by LDS
// zero-padding (no compares/selects in the hot loop).
// Row gather uses the gfx1250 async-to-LDS path (ASYNCcnt).
__global__ __launch_bounds__(128) void gate_net_kernel(
    const float* __restrict__ score,
    const int*   __restrict__ score_idx,
    float* __restrict__ out_fwd,
    float* __restrict__ out_bwd)
{
    __shared__ float s[PAD + N_SC + PAD];   // 768 floats = 3 KB

    const int d = blockIdx.x;
    const int t = threadIdx.x;              // 0..127

    // Zero the padding: s[0..255] and s[512..767].
    s[t]                   = 0.0f;
    s[128 + t]             = 0.0f;
    s[PAD + N_SC + t]      = 0.0f;
    s[PAD + N_SC + 128 + t]= 0.0f;

    // Gather sc[j] = score[score_idx[d, j+1]] into s[PAD + j], j = t and t+128,
    // via gfx1250 async-to-LDS (per-lane global addr -> per-lane LDS addr).
    {
        const size_t rowbase = (size_t)d * L_LEN + 1;
        const int idx0 = score_idx[rowbase + t];
        const int idx1 = score_idx[rowbase + t + 128];
        const uint64_t ga0 = (uint64_t)(score + idx0);
        const uint64_t ga1 = (uint64_t)(score + idx1);
        const uint32_t ld0 = (uint32_t)(uintptr_t)(&s[PAD + t]);
        const uint32_t ld1 = (uint32_t)(uintptr_t)(&s[PAD + t + 128]);
        asm volatile("global_load_async_to_lds_b32 %0, %1, off"
                     :: "v"(ld0), "v"(ga0) : "memory");
        asm volatile("global_load_async_to_lds_b32 %0, %1, off"
                     :: "v"(ld1), "v"(ga1) : "memory");
        asm volatile("s_wait_asynccnt 0" ::: "memory");
    }
    __syncthreads();

    const int j0 = 2 * t;                   // columns j0, j0+1
    const float* sf = &s[PAD + j0];         // fwd: sf[-r], sf[1-r]; bwd: sf[r], sf[r+1]

    // gate(h) = clamp((h - hj)*10 + 1, 0, 1); fold per-column constant c = 1 - 10*hj.
    const float c0 = fmaf(-10.0f, sf[0], 1.0f);
    const float c1 = fmaf(-10.0f, sf[1], 1.0f);

    float* fw = out_fwd + (size_t)d * N_R * N_SC + j0;
    float* bw = out_bwd + (size_t)d * N_R * N_SC + j0;

    float pf0 = 1.0f, pf1 = 1.0f, pb0 = 1.0f, pb1 = 1.0f;

#pragma unroll 5
    for (int r = 1; r <= N_R; ++r) {
        // Adjacent LDS reads -> ds_load_2addr_b32 pairs; padding supplies OOB zeros.
        const float f0 = sf[-r];
        const float f1 = sf[1 - r];
        const float b0 = sf[r];
        const float b1 = sf[r + 1];

        // clamp(fma) folds into v_fma_f32 ... clamp
        pf0 *= fminf(1.0f, fmaxf(0.0f, fmaf(f0, 10.0f, c0)));
        pf1 *= fminf(1.0f, fmaxf(0.0f, fmaf(f1, 10.0f, c1)));
        pb0 *= fminf(1.0f, fmaxf(0.0f, fmaf(b0, 10.0f, c0)));
        pb1 *= fminf(1.0f, fmaxf(0.0f, fmaf(b1, 10.0f, c1)));

        // Streaming b64 NT stores: write-once data, keep it out of L2.
        v2f vf = {pf0, pf1};
        v2f vb = {pb0, pb1};
        __builtin_nontemporal_store(vf, (v2f*)fw);
        __builtin_nontemporal_store(vb, (v2f*)bw);
        fw += N_SC;
        bw += N_SC;
    }
}

extern "C" void kernel_launch(void* const* d_in, const int* in_sizes, int n_in,
                              void* d_out, int out_size, void* d_ws, size_t ws_size,
                              hipStream_t stream) {
    const float* score     = (const float*)d_in[0];
    const int*   score_idx = (const int*)d_in[1];

    float* out     = (float*)d_out;
    float* out_fwd = out;                                   // (512, 255, 256)
    float* out_bwd = out + (size_t)D_ROWS * N_R * N_SC;     // (512, 255, 256)

    gate_net_kernel<<<dim3(D_ROWS), dim3(128), 0, stream>>>(
        score, score_idx, out_fwd, out_bwd);
}